// SoftmaxAttention_58626303590875
// MI455X (gfx1250) — compile-verified
//
#include <hip/hip_runtime.h>

// ESIM cross-attention as two fused flash-attention passes, fp32 WMMA
// (V_WMMA_F32_16X16X4_F32) on gfx1250, wave32.
// KV blocks are double-buffered in LDS and streamed with the Tensor Data
// Mover (TDM): the DMA of block kb+1 overlaps compute on block kb, with
// per-wave TENSORcnt ordering (s_wait_tensorcnt 1) + workgroup barriers.

typedef __attribute__((ext_vector_type(2))) float v2f;
typedef __attribute__((ext_vector_type(8))) float v8f;
typedef __attribute__((ext_vector_type(4))) unsigned int v4u;
typedef __attribute__((ext_vector_type(8))) int v8i;
typedef __attribute__((ext_vector_type(4))) int v4i;

#define SEQ 512
#define DIM 512
#define BQ  64
#define BK  32
#define NBLK (SEQ / BK)
#define NEG_MASK_F (-999999.0f)

// Issue a TDM load of a BK x DIM fp32 tile (contiguous rows, row stride DIM)
// from global memory into LDS at byte offset lds_off.
__device__ __forceinline__ void tdm_load_kv(const float* gptr, unsigned lds_off) {
  const unsigned long long ga = (unsigned long long)(uintptr_t)gptr;

  v4u g0;
  g0.x = 1u;                                     // count=1, valid user D#
  g0.y = lds_off;                                // lds_addr  [63:32]
  g0.z = (unsigned)(ga & 0xffffffffu);           // global_addr lo [95:64]
  g0.w = (unsigned)((ga >> 32) & 0x1ffffffu)     // global_addr hi [120:96]
         | (2u << 30);                           // type=2 ("image")

  v8i g1;
  g1[0] = 0x00020000;                            // wg_mask=0, data_size=2 (4B)
  g1[1] = (int)(512u << 16);                     // tensor_dim0[15:0]=512 @ [63:48]
  g1[2] = (int)(512u << 16);                     // dim0 hi=0 | tensor_dim1=512
  g1[3] = (int)(512u << 16);                     // dim1 hi=0 | tile_dim0=512
  g1[4] = BK;                                    // tile_dim1=BK, tile_dim2=0
  g1[5] = 512;                                   // tensor_dim0_stride = 512 elems
  g1[6] = 0;
  g1[7] = 0;

  v4i gz = {0, 0, 0, 0};                         // 2D tensor: groups 2/3 unused
#if defined(__clang_major__) && (__clang_major__ >= 23)
  v8i gz8 = {0, 0, 0, 0, 0, 0, 0, 0};
  __builtin_amdgcn_tensor_load_to_lds(g0, g1, gz, gz, gz8, 0);
#else
  __builtin_amdgcn_tensor_load_to_lds(g0, g1, gz, gz, 0);
#endif
}

__global__ __launch_bounds__(256, 1)
void esim_flash_attn_f32(const float* __restrict__ Q,
                         const float* __restrict__ KV,
                         const int*   __restrict__ qmask,
                         float* __restrict__ out) {
  __shared__ float sm_kv[2 * BK * DIM];  // 2 x 64 KB ping-pong KV buffers
  __shared__ float sm_s[BQ * BK];        // 8 KB: S tile, then P (= exp) tile
  __shared__ float sm_m[BQ];             // running row max
  __shared__ float sm_l[BQ];             // running row sum
  __shared__ float sm_a[BQ];             // per-block rescale factor alpha

  const int tid  = threadIdx.x;
  const int wave = tid >> 5;
  const int lane = tid & 31;
  const int lm   = lane & 15;   // N / M index within 16
  const int lh   = lane >> 4;   // half-wave select
  const int rw   = wave & 3;    // row group   (4 x 16 = 64 Q rows)
  const int cw   = wave >> 2;   // column half (S cols / 2 x 256 out cols)

  const int b  = blockIdx.x >> 3;
  const int q0 = (blockIdx.x & 7) * BQ;

  const float* Qb  = Q   + (size_t)b * SEQ * DIM;
  const float* KVb = KV  + (size_t)b * SEQ * DIM;
  float*       Ob  = out + (size_t)b * SEQ * DIM;
  const int*   qm  = qmask + b * SEQ;

  if (tid < BQ) { sm_m[tid] = -3.0e38f; sm_l[tid] = 0.0f; }

  // This wave's Q row for A-fragments (A 16x4 f32: lanes 0-15 K=0,1; 16-31 K=2,3)
  const float* qrow = Qb + (size_t)(q0 + rw * 16 + lm) * DIM;
  __builtin_prefetch(qrow, 0, 1);   // global_prefetch_b8: Q rows are L2-resident

  v8f acc[16];
  const v8f vzero = {0.f,0.f,0.f,0.f,0.f,0.f,0.f,0.f};
#pragma unroll
  for (int t = 0; t < 16; ++t) acc[t] = vzero;

  // ---- pipeline prologue: start DMA of block 0 into buffer 0
  if (wave == 0) {
    tdm_load_kv(KVb, (unsigned)(uintptr_t)(&sm_kv[0]));
  }

  for (int kb = 0; kb < NBLK; ++kb) {
    const float* kvb = &sm_kv[(kb & 1) * (BK * DIM)];

    // ---- issue DMA of block kb+1 into the other buffer, then wait for
    // block kb. TDM ops from one wave complete in order, so TENSORcnt<=1
    // means block kb has fully landed while kb+1 streams behind compute.
    if (wave == 0) {
      if (kb + 1 < NBLK) {
        tdm_load_kv(KVb + (size_t)(kb + 1) * BK * DIM,
                    (unsigned)(uintptr_t)(&sm_kv[((kb + 1) & 1) * (BK * DIM)]));
        __builtin_amdgcn_s_wait_tensorcnt(1);
      } else {
        __builtin_amdgcn_s_wait_tensorcnt(0);
      }
    }
    __syncthreads();   // publish buf[kb&1] to all waves

    // ---- phase 1: S = Q . KV^T  (this wave: one 16x16 tile of the 64x32 S)
    // Two independent accumulators (even/odd k-steps) keep WMMA ILP at 2.
    v8f sacc0 = vzero, sacc1 = vzero;
    const float* bcol = &kvb[(cw * 16 + lm) * DIM];
#pragma unroll 4
    for (int kk = 0; kk < DIM / 8; ++kk) {
      v2f a0 = *(const v2f*)(qrow + kk * 8 + 2 * lh);
      v2f b0 = *(const v2f*)(bcol + kk * 8 + 2 * lh);
      sacc0 = __builtin_amdgcn_wmma_f32_16x16x4_f32(false, a0, false, b0,
                                                    (short)0, sacc0, false, false);
      v2f a1 = *(const v2f*)(qrow + kk * 8 + 4 + 2 * lh);
      v2f b1 = *(const v2f*)(bcol + kk * 8 + 4 + 2 * lh);
      sacc1 = __builtin_amdgcn_wmma_f32_16x16x4_f32(false, a1, false, b1,
                                                    (short)0, sacc1, false, false);
    }
    // C/D layout: VGPR v -> row v (lanes 0-15) / v+8 (lanes 16-31), col = lm
#pragma unroll
    for (int v = 0; v < 8; ++v) {
      sm_s[(rw * 16 + v + 8 * lh) * BK + cw * 16 + lm] = sacc0[v] + sacc1[v];
    }
    __syncthreads();

    // ---- phase 2: online softmax update (row-constant mask term cancels via max)
    if (tid < BQ) {
      const int r = tid;
      const float cadd = (qm[q0 + r] == 0) ? NEG_MASK_F : 0.0f;
      const float mo = sm_m[r];
      float mx = mo;
      for (int c = 0; c < BK; ++c) mx = fmaxf(mx, sm_s[r * BK + c] + cadd);
      const float alpha = __expf(mo - mx);
      float sum = 0.0f;
      for (int c = 0; c < BK; ++c) {
        const float e = __expf(sm_s[r * BK + c] + cadd - mx);
        sm_s[r * BK + c] = e;
        sum += e;
      }
      sm_m[r] = mx;
      sm_l[r] = sm_l[r] * alpha + sum;
      sm_a[r] = alpha;
    }
    __syncthreads();

    // ---- phase 3: rescale accumulator, then acc += P . KV
    float al[8];
#pragma unroll
    for (int v = 0; v < 8; ++v) al[v] = sm_a[rw * 16 + v + 8 * lh];
#pragma unroll
    for (int t = 0; t < 16; ++t)
#pragma unroll
      for (int v = 0; v < 8; ++v) acc[t][v] *= al[v];

#pragma unroll 2
    for (int kk = 0; kk < BK / 4; ++kk) {
      // A 16x4 from P tile in LDS
      v2f a = *(const v2f*)(&sm_s[(rw * 16 + lm) * BK + kk * 4 + 2 * lh]);
      const int krow = kk * 4 + 2 * lh;
#pragma unroll
      for (int t = 0; t < 16; ++t) {
        const int dcol = cw * 256 + t * 16 + lm;
        v2f bb;
        bb.x = kvb[krow * DIM + dcol];
        bb.y = kvb[(krow + 1) * DIM + dcol];
        acc[t] = __builtin_amdgcn_wmma_f32_16x16x4_f32(false, a, false, bb,
                                                       (short)0, acc[t], false, false);
      }
    }
    // End-of-iteration barrier: guarantees nobody still reads the buffer the
    // TDM will overwrite at the top of the next iteration.
    __syncthreads();
  }

  // ---- epilogue: divide by softmax sum, store
  float lv[8];
#pragma unroll
  for (int v = 0; v < 8; ++v) lv[v] = 1.0f / sm_l[rw * 16 + v + 8 * lh];
#pragma unroll
  for (int t = 0; t < 16; ++t) {
    const int dcol = cw * 256 + t * 16 + lm;
#pragma unroll
    for (int v = 0; v < 8; ++v) {
      const int row = q0 + rw * 16 + v + 8 * lh;
      Ob[(size_t)row * DIM + dcol] = acc[t][v] * lv[v];
    }
  }
}

extern "C" void kernel_launch(void* const* d_in, const int* in_sizes, int n_in,
                              void* d_out, int out_size, void* d_ws, size_t ws_size,
                              hipStream_t stream) {
  const float* premise = (const float*)d_in[0];
  const int*   pmask   = (const int*)d_in[1];
  const float* hypo    = (const float*)d_in[2];
  const int*   hmask   = (const int*)d_in[3];
  float* out = (float*)d_out;

  const int B = 64;
  dim3 grid(B * (SEQ / BQ));  // 512 workgroups
  dim3 block(256);

  // Pass A: attended_premises = softmax_h(prem.hyp^T + rowconst) . hyp
  esim_flash_attn_f32<<<grid, block, 0, stream>>>(premise, hypo, pmask, out);
  // Pass B: attended_hypothesis = softmax_p(hyp.prem^T + rowconst) . prem
  esim_flash_attn_f32<<<grid, block, 0, stream>>>(hypo, premise, hmask,
                                                  out + (size_t)B * SEQ * DIM);
}